// GIN_73340861546937
// MI455X (gfx1250) — compile-verified
//
#include <hip/hip_runtime.h>
#include <hip/hip_bf16.h>

// ---------------------------------------------------------------------------
// GIN (3-layer) on MI455X / gfx1250 (wave32, WMMA).
//
// Pipeline per layer:
//   1) copy:      sum = x                     (agg includes self, eps=0)
//   2) scatter:   sum[dst] += x[src]          (global_atomic_add_f32, L2-resident)
//   3) wmma gemm: h = relu(BN((sum)@W1 + b1)) (V_WMMA_F32_16X16X4_F32, exact fp32)
//   4) wmma gemm: x = relu(h@W2 + b2)
// Then: pool (atomic scatter by graph id), wmma gemm for Wf1, scalar tail Wf2.
//
// Weights are pre-transposed once so A and B fragments are contiguous float2
// loads. BN folded to per-column scale/shift in a prep kernel.
// ---------------------------------------------------------------------------

typedef __attribute__((ext_vector_type(2))) float v2f;
typedef __attribute__((ext_vector_type(8))) float v8f;

#define GIN_N   100000
#define GIN_E   1638400
#define GIN_C   128
#define GIN_L   3
#define GIN_G   1000
#define GIN_OUT 10
#define LDSS    132   // LDS row stride in floats (avoids bank conflicts, keeps 8B align)

// ---- prep: transpose the 7 128x128 weight matrices into Wt[m][col][k] ------
__global__ void gin_transpose_w(const float* __restrict__ Ws1,
                                const float* __restrict__ Ws2,
                                const float* __restrict__ Wf1,
                                float* __restrict__ Wt) {
    int tid = blockIdx.x * blockDim.x + threadIdx.x;   // 7*128*128
    if (tid >= 7 * GIN_C * GIN_C) return;
    int m = tid >> 14;
    int rem = tid & 16383;
    int r = rem >> 7;      // k (row of W)
    int c = rem & 127;     // col of W
    const float* src = (m < 3) ? (Ws1 + m * GIN_C * GIN_C)
                     : (m < 6) ? (Ws2 + (m - 3) * GIN_C * GIN_C)
                               : Wf1;
    Wt[m * GIN_C * GIN_C + c * GIN_C + r] = src[r * GIN_C + c];
}

// ---- prep: fold BN(eval) into per-column scale/shift -----------------------
__global__ void gin_bn_prep(const float* __restrict__ gammas,
                            const float* __restrict__ betas,
                            const float* __restrict__ rms,
                            const float* __restrict__ rvs,
                            float* __restrict__ scale,
                            float* __restrict__ shift) {
    int tid = blockIdx.x * blockDim.x + threadIdx.x;   // L*C
    if (tid >= GIN_L * GIN_C) return;
    float s = gammas[tid] * rsqrtf(rvs[tid] + 1e-5f);
    scale[tid] = s;
    shift[tid] = betas[tid] - rms[tid] * s;
}

// ---- sum = x (float4 copy) -------------------------------------------------
__global__ void gin_copy(const float* __restrict__ src, float* __restrict__ dst) {
    int tid = blockIdx.x * blockDim.x + threadIdx.x;   // N*C/4
    if (tid >= GIN_N * GIN_C / 4) return;
    ((float4*)dst)[tid] = ((const float4*)src)[tid];
}

// ---- edge scatter: sum[dst] += x[src] --------------------------------------
__global__ void gin_scatter_edges(const float* __restrict__ x,
                                  const long long* __restrict__ ei,   // [2,E] int64
                                  float* __restrict__ sum) {
    long long tid = (long long)blockIdx.x * blockDim.x + threadIdx.x; // E*32
    if (tid >= (long long)GIN_E * 32) return;
    int e  = (int)(tid >> 5);
    int c4 = ((int)tid & 31) << 2;
    long long s = ei[e];
    long long d = ei[(long long)GIN_E + e];
    float4 v = *(const float4*)&x[s * GIN_C + c4];
    float* p = &sum[d * GIN_C + c4];
    atomicAdd(p + 0, v.x);
    atomicAdd(p + 1, v.y);
    atomicAdd(p + 2, v.z);
    atomicAdd(p + 3, v.w);
}

// ---- WMMA GEMM: out[M,128] = act( A[M,128] @ W[128,128] + b ) --------------
// Wt is W transposed: Wt[col][k]. One block = 16 rows x 128 cols, 8 waves,
// each wave owns a 16x16 tile, K swept 128 in steps of 4 (32 wmma ops).
__global__ void gin_wmma_gemm(const float* __restrict__ A,
                              const float* __restrict__ Wt,
                              const float* __restrict__ bias,
                              const float* __restrict__ scale,   // nullptr -> no BN
                              const float* __restrict__ shift,
                              float* __restrict__ out,
                              int M, int do_relu) {
    __shared__ float As[16 * LDSS];
    const int rowBase = blockIdx.x << 4;
    const int tid = threadIdx.x;

    // Stage the 16x128 A tile into LDS (guarded for the tail tile).
#pragma unroll
    for (int i = 0; i < 8; ++i) {
        int idx = tid + (i << 8);     // 256 threads * 8 = 2048 elements
        int r = idx >> 7;
        int c = idx & 127;
        int gr = rowBase + r;
        As[r * LDSS + c] = (gr < M) ? A[(size_t)gr * GIN_C + c] : 0.0f;
    }
    __syncthreads();

    const int wave = tid >> 5;
    const int lane = tid & 31;
    const int colBase = wave << 4;
    const int r    = lane & 15;            // A: row in tile; B/C/D: col in tile
    const int koff = (lane >> 4) << 1;     // upper half-wave holds K+2,K+3
    const int col  = colBase + r;          // global output column

    v8f acc = {};
#pragma unroll
    for (int k = 0; k < GIN_C; k += 4) {
        v2f a = *(const v2f*)&As[r * LDSS + k + koff];
        v2f b = *(const v2f*)&Wt[col * GIN_C + k + koff];
        // D = A(16x4) * B(4x16) + C, exact fp32
        acc = __builtin_amdgcn_wmma_f32_16x16x4_f32(
            false, a, false, b, (short)0, acc, false, false);
    }

    // Epilogue: bias (+BN scale/shift) (+ReLU), C/D layout:
    // vgpr i, lanes 0-15 -> M=i; lanes 16-31 -> M=i+8; N = lane&15.
    float bcol = bias[col];
    float scol = scale ? scale[col] : 1.0f;
    float tcol = shift ? shift[col] : 0.0f;
    int rbase2 = rowBase + ((lane >> 4) << 3);
#pragma unroll
    for (int i = 0; i < 8; ++i) {
        int grow = rbase2 + i;
        if (grow < M) {
            float v = acc[i] + bcol;
            if (scale) v = v * scol + tcol;
            if (do_relu) v = fmaxf(v, 0.0f);
            out[(size_t)grow * GIN_C + col] = v;
        }
    }
}

// ---- zero a float buffer ---------------------------------------------------
__global__ void gin_zero(float* __restrict__ p, int n) {
    int tid = blockIdx.x * blockDim.x + threadIdx.x;
    if (tid < n) p[tid] = 0.0f;
}

// ---- global_add_pool: g[batch[i]] += x[i] ----------------------------------
__global__ void gin_pool_scatter(const float* __restrict__ x,
                                 const long long* __restrict__ batch,
                                 float* __restrict__ g) {
    long long tid = (long long)blockIdx.x * blockDim.x + threadIdx.x; // N*32
    if (tid >= (long long)GIN_N * 32) return;
    int i  = (int)(tid >> 5);
    int c4 = ((int)tid & 31) << 2;
    long long b = batch[i];
    float4 v = *(const float4*)&x[(size_t)i * GIN_C + c4];
    float* p = &g[b * GIN_C + c4];
    atomicAdd(p + 0, v.x);
    atomicAdd(p + 1, v.y);
    atomicAdd(p + 2, v.z);
    atomicAdd(p + 3, v.w);
}

// ---- final tiny GEMM: out[G,10] = gh[G,128] @ Wf2[128,10] + bf2 ------------
__global__ void gin_final_fc(const float* __restrict__ gh,
                             const float* __restrict__ Wf2,
                             const float* __restrict__ bf2,
                             float* __restrict__ out) {
    int tid = blockIdx.x * blockDim.x + threadIdx.x;   // G*OUT
    if (tid >= GIN_G * GIN_OUT) return;
    int g = tid / GIN_OUT;
    int o = tid % GIN_OUT;
    float s = bf2[o];
#pragma unroll 8
    for (int k = 0; k < GIN_C; ++k)
        s += gh[g * GIN_C + k] * Wf2[k * GIN_OUT + o];
    out[tid] = s;
}

// ---------------------------------------------------------------------------
extern "C" void kernel_launch(void* const* d_in, const int* in_sizes, int n_in,
                              void* d_out, int out_size, void* d_ws, size_t ws_size,
                              hipStream_t stream) {
    (void)in_sizes; (void)n_in; (void)out_size; (void)ws_size;

    const float*     x      = (const float*)d_in[0];
    const long long* ei     = (const long long*)d_in[1];   // int64 [2,E]
    const long long* batch  = (const long long*)d_in[2];   // int64 [N]
    const float*     Ws1    = (const float*)d_in[4];
    const float*     bs1    = (const float*)d_in[5];
    const float*     gammas = (const float*)d_in[6];
    const float*     betas  = (const float*)d_in[7];
    const float*     rms    = (const float*)d_in[8];
    const float*     rvs    = (const float*)d_in[9];
    const float*     Ws2    = (const float*)d_in[10];
    const float*     bs2    = (const float*)d_in[11];
    const float*     Wf1    = (const float*)d_in[12];
    const float*     bf1    = (const float*)d_in[13];
    const float*     Wf2    = (const float*)d_in[14];
    const float*     bf2    = (const float*)d_in[15];
    float*           outp   = (float*)d_out;

    const size_t NC = (size_t)GIN_N * GIN_C;
    float* w     = (float*)d_ws;
    float* B0    = w;                       // [N,C]
    float* B1    = B0 + NC;                 // [N,C]
    float* Wt    = B1 + NC;                 // 7 * 128*128
    float* scl   = Wt + 7 * GIN_C * GIN_C;  // [L,C]
    float* shf   = scl + GIN_L * GIN_C;     // [L,C]
    float* gbuf  = shf + GIN_L * GIN_C;     // [G,C]
    float* ghbuf = gbuf + GIN_G * GIN_C;    // [G,C]

    // ---- one-time prep (cheap, re-run every call for determinism) ----
    gin_transpose_w<<<(7 * GIN_C * GIN_C + 255) / 256, 256, 0, stream>>>(Ws1, Ws2, Wf1, Wt);
    gin_bn_prep<<<(GIN_L * GIN_C + 127) / 128, 128, 0, stream>>>(gammas, betas, rms, rvs, scl, shf);

    // ---- 3 GIN layers (ping-pong B0/B1) ----
    const int copyBlocks    = (int)(NC / 4 / 256);                 // 12500
    const int scatterBlocks = (int)((long long)GIN_E * 32 / 256);  // 204800
    const int gemmBlocksN   = GIN_N / 16;                          // 6250

    const float* cur = x;
    for (int l = 0; l < GIN_L; ++l) {
        float* sbuf = (l & 1) ? B1 : B0;
        float* mbuf = (l & 1) ? B0 : B1;

        gin_copy<<<copyBlocks, 256, 0, stream>>>(cur, sbuf);
        gin_scatter_edges<<<scatterBlocks, 256, 0, stream>>>(cur, ei, sbuf);

        gin_wmma_gemm<<<gemmBlocksN, 256, 0, stream>>>(
            sbuf, Wt + l * GIN_C * GIN_C, bs1 + l * GIN_C,
            scl + l * GIN_C, shf + l * GIN_C, mbuf, GIN_N, 1);

        gin_wmma_gemm<<<gemmBlocksN, 256, 0, stream>>>(
            mbuf, Wt + (3 + l) * GIN_C * GIN_C, bs2 + l * GIN_C,
            nullptr, nullptr, sbuf, GIN_N, 1);

        cur = sbuf;
    }

    // ---- global_add_pool ----
    gin_zero<<<(GIN_G * GIN_C + 255) / 256, 256, 0, stream>>>(gbuf, GIN_G * GIN_C);
    gin_pool_scatter<<<(int)((long long)GIN_N * 32 / 256), 256, 0, stream>>>(cur, batch, gbuf);

    // ---- final MLP: relu(g @ Wf1 + bf1) via WMMA, then small FC ----
    gin_wmma_gemm<<<(GIN_G + 15) / 16, 256, 0, stream>>>(
        gbuf, Wt + 6 * GIN_C * GIN_C, bf1, nullptr, nullptr, ghbuf, GIN_G, 1);

    gin_final_fc<<<(GIN_G * GIN_OUT + 255) / 256, 256, 0, stream>>>(ghbuf, Wf2, bf2, outp);
}